// GCNModel_48687749267591
// MI455X (gfx1250) — compile-verified
//
#include <hip/hip_runtime.h>

typedef float v2f __attribute__((ext_vector_type(2)));
typedef float v8f __attribute__((ext_vector_type(8)));
typedef unsigned int u32x4 __attribute__((ext_vector_type(4)));
typedef int i32x4 __attribute__((ext_vector_type(4)));
typedef int i32x8 __attribute__((ext_vector_type(8)));

#define N_NODES 50000
#define N_EDGES 1200000
#define IN_FEATS 256
#define HIDDEN 64
#define OUT_F 32
#define M_TILES (N_NODES / 16)   // 3125, exact
#define BLOCK_T 256

// ---------------- init: zero degree arrays and aggregation buffer ----------------
__global__ void gcn_zero_kernel(float* __restrict__ deg_out, float* __restrict__ deg_in,
                                float* __restrict__ agg) {
    int i = blockIdx.x * BLOCK_T + threadIdx.x;
    if (i < N_NODES * HIDDEN) agg[i] = 0.0f;
    if (i < N_NODES) { deg_out[i] = 0.0f; deg_in[i] = 0.0f; }
}

// ---------------- degree accumulation (L2 atomics) ----------------
__global__ void gcn_degree_kernel(const int* __restrict__ src, const int* __restrict__ dst,
                                  float* __restrict__ deg_out, float* __restrict__ deg_in) {
    int e = blockIdx.x * BLOCK_T + threadIdx.x;
    if (e >= N_EDGES) return;
    atomicAdd(&deg_out[src[e]], 1.0f);
    atomicAdd(&deg_in[dst[e]], 1.0f);
}

// ---------------- norm = clip(deg,1)^-0.5, in place ----------------
__global__ void gcn_norm_kernel(float* __restrict__ deg_out, float* __restrict__ deg_in) {
    int i = blockIdx.x * BLOCK_T + threadIdx.x;
    if (i >= N_NODES) return;
    deg_out[i] = rsqrtf(fmaxf(deg_out[i], 1.0f));
    deg_in[i]  = rsqrtf(fmaxf(deg_in[i], 1.0f));
}

// ---------------- GEMM1: h[N,64] = n_feat[N,256] @ W_gcn[256,64] via WMMA f32 ----------------
// One wave computes a 16-row tile x all 64 cols (4 accumulators).
// W is staged in LDS in 128-K chunks by the Tensor Data Mover; the D# pad feature
// (pad 1 dword every 64 dwords) produces the bank-conflict-free stride-65 layout.
#define G1_CHUNK_K 128
#define G1_LDS_STRIDE 65
__global__ __launch_bounds__(BLOCK_T)
void gcn_gemm1_wmma(const float* __restrict__ A, const float* __restrict__ W,
                    float* __restrict__ H) {
    __shared__ float ldsW[G1_CHUNK_K * G1_LDS_STRIDE]; // 33,280 B

    const int wave = threadIdx.x >> 5;
    const int lane = threadIdx.x & 31;
    const int half = lane >> 4;      // 0: K pair {0,1}; 1: K pair {2,3}
    const int l16  = lane & 15;
    const int rowTile = blockIdx.x * 8 + wave;
    const bool active = (rowTile < M_TILES);
    const int m0 = rowTile * 16;
    const float* arow = A + (size_t)(m0 + l16) * IN_FEATS;

    // LDS byte offset of ldsW (addrspace(3) pointer value == LDS offset)
    const unsigned ldsOff =
        (unsigned)(unsigned long long)(__attribute__((address_space(3))) float*)ldsW;

    v8f acc[4] = {};

    for (int kc = 0; kc < IN_FEATS; kc += G1_CHUNK_K) {
        __syncthreads();   // previous chunk fully consumed before TDM overwrites LDS
        if (wave == 0) {
            const unsigned long long gaddr =
                (unsigned long long)(const void*)(W + (size_t)kc * HIDDEN);
            const unsigned nelem = G1_CHUNK_K * HIDDEN;   // 8192 f32, contiguous

            u32x4 g0;
            g0[0] = 1u;                                    // count=1 (valid user D#)
            g0[1] = ldsOff;                                // lds_addr (bytes)
            g0[2] = (unsigned)(gaddr & 0xFFFFFFFFull);     // global_addr[31:0]
            g0[3] = (unsigned)((gaddr >> 32) & 0x1FFFFFFull) // global_addr[56:32]
                    | (2u << 30);                          // type=2 ("image")

            i32x8 g1;
            g1[0] = (int)((2u << 16)      // data_size = 4 bytes
                    | (1u << 20)          // pad_enable
                    | (5u << 22));        // pad_interval: every 64 dwords
                                          // pad_amount = 0 -> 1 dword
            g1[1] = (int)((nelem & 0xFFFFu) << 16);  // tensor_dim0[15:0] @ bits63:48
            g1[2] = (int)(1u << 16);      // tensor_dim0[31:16]=0 | tensor_dim1=1
            g1[3] = (int)((nelem & 0xFFFFu) << 16);  // tensor_dim1 hi=0 | tile_dim0
            g1[4] = 1;                    // tile_dim1=1, tile_dim2=0
            g1[5] = (int)nelem;           // tensor_dim0_stride[31:0]
            g1[6] = 0;                    // stride0[47:32]=0 | stride1 lo=0
            g1[7] = 0;                    // stride1 hi

            i32x4 g2 = {0, 0, 0, 0};      // <=2D tensor: groups 2/3 unused
            i32x4 g3 = {0, 0, 0, 0};
            i32x8 g4 = {0, 0, 0, 0, 0, 0, 0, 0};  // 6-arg toolchain: trailing group, unused
            __builtin_amdgcn_tensor_load_to_lds(g0, g1, g2, g3, g4, 0);
            __builtin_amdgcn_s_wait_tensorcnt((short)0);
        }
        __syncthreads();   // LDS chunk visible to all waves

        if (active) {
            for (int k0 = 0; k0 < G1_CHUNK_K; k0 += 4) {
                const int kk = k0 + 2 * half;
                v2f a = *(const v2f*)(arow + kc + kk);   // A[m0+l16][kc+kk .. +1]
#pragma unroll
                for (int t = 0; t < 4; ++t) {
                    const int n = t * 16 + l16;
                    v2f b;
                    b[0] = ldsW[kk * G1_LDS_STRIDE + n];         // B[kk][n]
                    b[1] = ldsW[(kk + 1) * G1_LDS_STRIDE + n];   // B[kk+1][n]
                    acc[t] = __builtin_amdgcn_wmma_f32_16x16x4_f32(
                        false, a, false, b, (short)0, acc[t], false, false);
                }
            }
        }
    }

    if (active) {
#pragma unroll
        for (int t = 0; t < 4; ++t)
#pragma unroll
            for (int r = 0; r < 8; ++r)
                H[(size_t)(m0 + r + 8 * half) * HIDDEN + t * 16 + l16] = acc[t][r];
    }
}

// ---------------- edge scatter: agg[dst] += h[src] * norm_src[src] ----------------
// One thread per (edge, feature): coalesced h-row reads, f32 atomics land in L2.
__global__ __launch_bounds__(BLOCK_T)
void gcn_scatter_kernel(const int* __restrict__ src, const int* __restrict__ dst,
                        const float* __restrict__ h, const float* __restrict__ norm_src,
                        float* __restrict__ agg) {
    unsigned int tid = blockIdx.x * BLOCK_T + threadIdx.x;
    int e = (int)(tid >> 6);
    if (e >= N_EDGES) return;
    int j = (int)(tid & 63);
    int s = src[e];
    int d = dst[e];
    float v = h[(size_t)s * HIDDEN + j] * norm_src[s];
    atomicAdd(&agg[(size_t)d * HIDDEN + j], v);
}

// ---------------- x = relu(agg * norm_dst + b_gcn), in place ----------------
__global__ void gcn_relu_kernel(float* __restrict__ agg, const float* __restrict__ norm_dst,
                                const float* __restrict__ b) {
    int i = blockIdx.x * BLOCK_T + threadIdx.x;
    if (i >= N_NODES * HIDDEN) return;
    int node = i >> 6, j = i & 63;
    float x = fmaf(agg[i], norm_dst[node], b[j]);
    agg[i] = fmaxf(x, 0.0f);
}

// ---------------- GEMM2: out[N,32] = x[N,64] @ W_lin[64,32] + b_lin via WMMA f32 ----------------
#define G2_LDS_STRIDE 33
__global__ __launch_bounds__(BLOCK_T)
void gcn_gemm2_wmma(const float* __restrict__ X, const float* __restrict__ W,
                    const float* __restrict__ bias, float* __restrict__ Out) {
    __shared__ float ldsW[HIDDEN * G2_LDS_STRIDE]; // 8,448 B

    for (int idx = threadIdx.x; idx < HIDDEN * OUT_F; idx += BLOCK_T) {
        int r = idx >> 5, c = idx & 31;
        ldsW[r * G2_LDS_STRIDE + c] = W[idx];
    }
    __syncthreads();

    const int wave = threadIdx.x >> 5;
    const int lane = threadIdx.x & 31;
    const int half = lane >> 4;
    const int l16  = lane & 15;
    const int rowTile = blockIdx.x * 8 + wave;
    if (rowTile >= M_TILES) return;
    const int m0 = rowTile * 16;
    const float* xrow = X + (size_t)(m0 + l16) * HIDDEN;

    v8f acc[2] = {};
    for (int k0 = 0; k0 < HIDDEN; k0 += 4) {
        const int kk = k0 + 2 * half;
        v2f a = *(const v2f*)(xrow + kk);
#pragma unroll
        for (int t = 0; t < 2; ++t) {
            const int n = t * 16 + l16;
            v2f b;
            b[0] = ldsW[kk * G2_LDS_STRIDE + n];
            b[1] = ldsW[(kk + 1) * G2_LDS_STRIDE + n];
            acc[t] = __builtin_amdgcn_wmma_f32_16x16x4_f32(
                false, a, false, b, (short)0, acc[t], false, false);
        }
    }

#pragma unroll
    for (int t = 0; t < 2; ++t) {
        float bl = bias[t * 16 + l16];
#pragma unroll
        for (int r = 0; r < 8; ++r)
            Out[(size_t)(m0 + r + 8 * half) * OUT_F + t * 16 + l16] = acc[t][r] + bl;
    }
}

extern "C" void kernel_launch(void* const* d_in, const int* in_sizes, int n_in,
                              void* d_out, int out_size, void* d_ws, size_t ws_size,
                              hipStream_t stream) {
    const int*   src    = (const int*)d_in[0];
    const int*   dst    = (const int*)d_in[1];
    const float* n_feat = (const float*)d_in[2];
    const float* W_gcn  = (const float*)d_in[3];
    const float* b_gcn  = (const float*)d_in[4];
    const float* W_lin  = (const float*)d_in[5];
    const float* b_lin  = (const float*)d_in[6];
    float* out = (float*)d_out;

    // workspace layout: deg_out/norm_src [N], deg_in/norm_dst [N], h [N*64], agg [N*64]
    float* deg_out = (float*)d_ws;
    float* deg_in  = deg_out + N_NODES;
    float* h       = deg_in  + N_NODES;
    float* agg     = h + (size_t)N_NODES * HIDDEN;

    const int T = BLOCK_T;
    const int nh_blocks   = (N_NODES * HIDDEN + T - 1) / T;          // 12,500
    const int edge_blocks = (N_EDGES + T - 1) / T;                    // 4,688
    const int node_blocks = (N_NODES + T - 1) / T;                    // 196
    const int gemm_blocks = (M_TILES + 7) / 8;                        // 391 (8 waves/block)
    const int scat_blocks = (N_EDGES * 64) / T;                       // 300,000 exact

    gcn_zero_kernel<<<nh_blocks, T, 0, stream>>>(deg_out, deg_in, agg);
    gcn_degree_kernel<<<edge_blocks, T, 0, stream>>>(src, dst, deg_out, deg_in);
    gcn_norm_kernel<<<node_blocks, T, 0, stream>>>(deg_out, deg_in);
    gcn_gemm1_wmma<<<gemm_blocks, T, 0, stream>>>(n_feat, W_gcn, h);
    gcn_scatter_kernel<<<scat_blocks, T, 0, stream>>>(src, dst, h, deg_out, agg);
    gcn_relu_kernel<<<nh_blocks, T, 0, stream>>>(agg, deg_in, b_gcn);
    gcn_gemm2_wmma<<<gemm_blocks, T, 0, stream>>>(agg, W_lin, b_lin, out);
}